// GraphConv_32607391711589
// MI455X (gfx1250) — compile-verified
//
#include <hip/hip_runtime.h>
#include <stdint.h>

// CDNA5 / gfx1250, wave32.
typedef __attribute__((ext_vector_type(2))) float v2f;
typedef __attribute__((ext_vector_type(8))) float v8f;
typedef __attribute__((ext_vector_type(4))) unsigned v4u;
typedef __attribute__((ext_vector_type(8))) unsigned v8u;

#define N_NODES   100000
#define DIM       256
#define N_EDGES   1600000
#define LDS_STRIDE 260   // 256 + 4 pad: row*260 % 64 = 4*row -> conflict-free col reads

#if defined(__gfx1250__)
#define USE_TDM 1
#else
#define USE_TDM 0
#endif

// ---------------------------------------------------------------------------
// K1: agg = feat   (float4 streaming copy; agg lives in workspace)
// ---------------------------------------------------------------------------
__global__ void __launch_bounds__(256)
copy_feat_kernel(const float* __restrict__ feat, float* __restrict__ agg) {
    size_t i = (size_t)blockIdx.x * blockDim.x + threadIdx.x;   // float4 index
    const float4* s = (const float4*)feat;
    float4*       d = (float4*)agg;
    d[i] = s[i];
}

// ---------------------------------------------------------------------------
// K2: agg[dst[e]] += feat[src[e]]   (64 lanes per edge, float4 gather,
//     4x global_atomic_add_f32 per lane; feat & agg both fit in 192MB L2)
// ---------------------------------------------------------------------------
__global__ void __launch_bounds__(256)
scatter_add_kernel(const float* __restrict__ feat,
                   const int*   __restrict__ src,
                   const int*   __restrict__ dst,
                   float*       __restrict__ agg) {
    int e   = blockIdx.x * 4 + (threadIdx.x >> 6);   // 4 edges per block
    int col = (threadIdx.x & 63) * 4;
    int s = src[e];
    int d = dst[e];
    const float4 v = *(const float4*)(feat + (size_t)s * DIM + col);
    float* base = agg + (size_t)d * DIM + col;
    atomicAdd(base + 0, v.x);
    atomicAdd(base + 1, v.y);
    atomicAdd(base + 2, v.z);
    atomicAdd(base + 3, v.w);
}

// ---------------------------------------------------------------------------
// K3: out = relu(A @ W), fp32 WMMA 16x16x4.
// Block = 256 threads = 8 waves. The 32x256 A slab is staged into LDS by a
// single Tensor Data Mover descriptor (TENSOR_LOAD_TO_LDS, TENSORcnt),
// issued by wave 0, with TDM padding inserting 4 DWORDs every 256 DWORDs so
// the LDS image lands at the bank-conflict-free 260-float row stride.
//   wave w -> both 16-row M-tiles of the slab, N-tiles {2w, 2w+1}
// -> 4 WMMAs per k-step against 4 scalar W loads (B shared by the M-tiles);
// A fragments come from LDS as ds_load_2addr_b64.
// A-fragment layout (ISA 16x4 f32): lanes 0-15 carry K{+0,+1}, lanes 16-31
// carry K{+2,+3}; B mirrors that with N striped across the 16-lane halves.
// ---------------------------------------------------------------------------
__global__ void __launch_bounds__(256)
gemm_relu_kernel(const float* __restrict__ A,   // [N_NODES x 256] = agg + feat
                 const float* __restrict__ W,   // [256 x 256] row-major (K x N)
                 float*       __restrict__ out) {
    __shared__ float As[32 * LDS_STRIDE];

    const int mbase = blockIdx.x * 32;

    // Keep W hot in L2: each thread prefetches one W row's cacheline.
    __builtin_prefetch(W + ((size_t)threadIdx.x << 8), 0, 1);

#if USE_TDM
    if (threadIdx.x < 32) {   // wave 0 issues the DMA (TDM ignores EXEC)
        // Low 32 bits of a flat LDS-aperture pointer are the LDS byte offset
        // (CDNA5 ISA 10.2).
        const unsigned lds_base = (unsigned)(uintptr_t)(void*)&As[0];
        const unsigned long long gaddr =
            (unsigned long long)(uintptr_t)(A + (size_t)mbase * DIM);

        // ---- D# group 0 (ISA 8.3): count=1, lds_addr, global_addr, type=2
        v4u g0;
        g0.x = 1u;                                   // count=1, user mode
        g0.y = lds_base;                             // lds_addr [63:32]
        g0.z = (unsigned)gaddr;                      // global_addr [95:64]
        g0.w = ((unsigned)(gaddr >> 32) & 0x01FFFFFFu) | (2u << 30); // [120:96] + type=2

        // ---- D# group 1 (ISA 8.4): 2D tile 256 x 32, f32, LDS padding
        // data_size=2 (4B)          -> bits [17:16]
        // pad_enable=1              -> bit  [20]
        // pad_interval=7 (256 DW)   -> bits [24:22]
        // pad_amount=3 (4 DW)       -> bits [31:25]
        v8u g1;
        g1[0] = (2u << 16) | (1u << 20) | (7u << 22) | (3u << 25);
        g1[1] = (unsigned)(DIM) << 16;       // tensor_dim0[15:0] in [63:48]
        g1[2] = 32u << 16;                   // tensor_dim0 hi=0, tensor_dim1[15:0]=32
        g1[3] = (unsigned)(DIM) << 16;       // tensor_dim1 hi=0, tile_dim0=256
        g1[4] = 32u;                         // tile_dim1=32, tile_dim2=0
        g1[5] = (unsigned)(DIM);             // tensor_dim0_stride lo = 256
        g1[6] = 0u;                          // stride hi, tensor_dim1_stride lo
        g1[7] = 0u;                          // tensor_dim1_stride hi

        asm volatile("tensor_load_to_lds %0, %1" :: "s"(g0), "s"(g1) : "memory");
        __builtin_amdgcn_s_wait_tensorcnt(0);
    }
#else
    for (int f = threadIdx.x; f < 32 * 64; f += 256) {
        int row  = f >> 6;
        int colv = f & 63;
        const float4 v = *(const float4*)(A + (size_t)(mbase + row) * DIM + colv * 4);
        *(float4*)(&As[row * LDS_STRIDE + colv * 4]) = v;
    }
#endif
    __syncthreads();

    const int wave = threadIdx.x >> 5;
    const int lane = threadIdx.x & 31;
    const int l16  = lane & 15;
    const int hi   = lane >> 4;      // 0: lanes 0-15, 1: lanes 16-31
    const int koff = hi * 2;         // K sub-offset carried by this lane half

    v8f acc00 = {}, acc01 = {}, acc10 = {}, acc11 = {};

    const float* arow0 = &As[l16 * LDS_STRIDE];          // M-tile 0
    const float* arow1 = &As[(16 + l16) * LDS_STRIDE];   // M-tile 1
    const int    nb    = wave * 32;
    const int    ncol0 = nb + l16;                       // N-tile 0
    const int    ncol1 = nb + 16 + l16;                  // N-tile 1

    #pragma unroll 4
    for (int kb = 0; kb < DIM; kb += 4) {
        v2f a0, a1;
        a0.x = arow0[kb + koff];
        a0.y = arow0[kb + koff + 1];
        a1.x = arow1[kb + koff];
        a1.y = arow1[kb + koff + 1];

        const float* w0 = W + (size_t)(kb + koff) * DIM;   // row kb+koff
        const float* w1 = w0 + DIM;                        // row kb+koff+1
        v2f b0, b1;
        b0.x = w0[ncol0]; b0.y = w1[ncol0];
        b1.x = w0[ncol1]; b1.y = w1[ncol1];

        acc00 = __builtin_amdgcn_wmma_f32_16x16x4_f32(false, a0, false, b0,
                                                      (short)0, acc00, false, false);
        acc10 = __builtin_amdgcn_wmma_f32_16x16x4_f32(false, a1, false, b0,
                                                      (short)0, acc10, false, false);
        acc01 = __builtin_amdgcn_wmma_f32_16x16x4_f32(false, a0, false, b1,
                                                      (short)0, acc01, false, false);
        acc11 = __builtin_amdgcn_wmma_f32_16x16x4_f32(false, a1, false, b1,
                                                      (short)0, acc11, false, false);
    }

    // D layout: VGPR r -> row (r + hi*8), col l16 within each 16x16 tile.
    const int mout0 = mbase + hi * 8;        // M-tile 0 rows
    const int mout1 = mbase + 16 + hi * 8;   // M-tile 1 rows
    #pragma unroll
    for (int r = 0; r < 8; ++r) {
        float* o0 = out + (size_t)(mout0 + r) * DIM;
        float* o1 = out + (size_t)(mout1 + r) * DIM;
        float x00 = acc00[r], x01 = acc01[r], x10 = acc10[r], x11 = acc11[r];
        o0[ncol0] = x00 > 0.0f ? x00 : 0.0f;
        o0[ncol1] = x01 > 0.0f ? x01 : 0.0f;
        o1[ncol0] = x10 > 0.0f ? x10 : 0.0f;
        o1[ncol1] = x11 > 0.0f ? x11 : 0.0f;
    }
}

// ---------------------------------------------------------------------------
extern "C" void kernel_launch(void* const* d_in, const int* in_sizes, int n_in,
                              void* d_out, int out_size, void* d_ws, size_t ws_size,
                              hipStream_t stream) {
    const float* feat = (const float*)d_in[0];   // [100000, 256] f32
    const float* w1   = (const float*)d_in[1];   // [256, 256] f32
    const int*   src  = (const int*)d_in[2];     // [1600000] indices
    const int*   dst  = (const int*)d_in[3];     // [1600000] indices
    float*       out  = (float*)d_out;           // [100000, 256] f32
    float*       agg  = (float*)d_ws;            // [100000, 256] f32 scratch

    (void)in_sizes; (void)n_in; (void)out_size; (void)ws_size;

    // agg = feat  (so agg ends up holding agg_edges + feat)
    copy_feat_kernel<<<(N_NODES * DIM / 4) / 256, 256, 0, stream>>>(feat, agg);

    // agg[dst[e]] += feat[src[e]]  (4 edges per 256-thread block)
    scatter_add_kernel<<<N_EDGES / 4, 256, 0, stream>>>(feat, src, dst, agg);

    // out = relu(agg @ W)  (100000 = 3125 * 32, exact)
    gemm_relu_kernel<<<N_NODES / 32, 256, 0, stream>>>(agg, w1, out);
}